// Model_41025527611968
// MI455X (gfx1250) — compile-verified
//
#include <hip/hip_runtime.h>
#include <math.h>

#define Bn 256
#define NVV 10475
#define NKK 144
#define MVV 778
#define MKK 21
#define NBB 8192
#define KNN 9

typedef __attribute__((ext_vector_type(2))) float v2f;
typedef __attribute__((ext_vector_type(8))) float v8f;

// ---------------------------------------------------------------------------
// wave32 butterfly reduction (all lanes get the sum)
__device__ inline float wred(float v) {
    #pragma unroll
    for (int off = 16; off > 0; off >>= 1) v += __shfl_xor(v, off, 32);
    return v;
}

// ---------------------------------------------------------------------------
// Kernel 1: per (batch, hand) compute fused affine transform M (3x3) + T (3)
//   out_vert[n] = vin[n] @ (R_hand @ R_align) + T
// Stored in ws: MT[(hand*Bn+b)*12 + 0..8] = M row-major, +9..11 = T
__global__ void xform_kernel(const float* __restrict__ smplx_kpt,
                             const float* __restrict__ rkpt,
                             const float* __restrict__ rpose,
                             const float* __restrict__ lkpt,
                             const float* __restrict__ lpose,
                             const int* __restrict__ rigid_align_idx,
                             const int* __restrict__ r_align_smplx,
                             const int* __restrict__ l_align_smplx,
                             float* __restrict__ MT) {
    const int b = blockIdx.x;
    const int hand = blockIdx.y;
    const int lane = threadIdx.x;
    const float* kpt  = hand ? lkpt  : rkpt;
    const float* pose = hand ? lpose : rpose;
    const int*   asmx = hand ? l_align_smplx : r_align_smplx;

    // Rodrigues (redundant across lanes)
    float ax = pose[b * 3 + 0], ay = pose[b * 3 + 1], az = pose[b * 3 + 2];
    float ang = sqrtf(ax * ax + ay * ay + az * az);
    float inv = 1.0f / (ang + 1e-8f);
    float x = ax * inv, y = ay * inv, z = az * inv;
    float c = cosf(ang), s = sinf(ang), Cc = 1.0f - c;
    float Rh[3][3];
    Rh[0][0] = c + x * x * Cc; Rh[0][1] = x * y * Cc - z * s; Rh[0][2] = x * z * Cc + y * s;
    Rh[1][0] = y * x * Cc + z * s; Rh[1][1] = c + y * y * Cc; Rh[1][2] = y * z * Cc - x * s;
    Rh[2][0] = z * x * Cc - y * s; Rh[2][1] = z * y * Cc + x * s; Rh[2][2] = c + z * z * Cc;

    // per-lane keypoint: X[n] = Rh^T @ kpt[rigid_align_idx[n]], Y[n] = smplx_kpt[asmx[n]]
    float act = (lane < MKK) ? 1.0f : 0.0f;
    int ln = (lane < MKK) ? lane : 0;
    int m = rigid_align_idx[ln];
    const float* kp = kpt + ((size_t)b * MKK + m) * 3;
    float k0 = kp[0], k1 = kp[1], k2 = kp[2];
    float X0 = Rh[0][0] * k0 + Rh[1][0] * k1 + Rh[2][0] * k2;
    float X1 = Rh[0][1] * k0 + Rh[1][1] * k1 + Rh[2][1] * k2;
    float X2 = Rh[0][2] * k0 + Rh[1][2] * k1 + Rh[2][2] * k2;
    int yi = asmx[ln];
    const float* yp = smplx_kpt + ((size_t)b * NKK + yi) * 3;
    float Y0 = yp[0], Y1 = yp[1], Y2 = yp[2];

    const float invn = 1.0f / (float)MKK;
    float muX0 = wred(X0 * act) * invn, muX1 = wred(X1 * act) * invn, muX2 = wred(X2 * act) * invn;
    float muY0 = wred(Y0 * act) * invn, muY1 = wred(Y1 * act) * invn, muY2 = wred(Y2 * act) * invn;

    float dX0 = (X0 - muX0) * act, dX1 = (X1 - muX1) * act, dX2 = (X2 - muX2) * act;
    float dY0 = Y0 - muY0, dY1 = Y1 - muY1, dY2 = Y2 - muY2;

    float C_[3][3];
    C_[0][0] = wred(dX0 * dY0); C_[0][1] = wred(dX0 * dY1); C_[0][2] = wred(dX0 * dY2);
    C_[1][0] = wred(dX1 * dY0); C_[1][1] = wred(dX1 * dY1); C_[1][2] = wred(dX1 * dY2);
    C_[2][0] = wred(dX2 * dY0); C_[2][1] = wred(dX2 * dY1); C_[2][2] = wred(dX2 * dY2);

    // Jacobi eigendecomposition of S = C^T C  (redundant per lane)
    float S[3][3], V[3][3];
    #pragma unroll
    for (int i = 0; i < 3; ++i)
        #pragma unroll
        for (int j = 0; j < 3; ++j) {
            S[i][j] = C_[0][i] * C_[0][j] + C_[1][i] * C_[1][j] + C_[2][i] * C_[2][j];
            V[i][j] = (i == j) ? 1.0f : 0.0f;
        }
    for (int sweep = 0; sweep < 10; ++sweep) {
        #pragma unroll
        for (int pi = 0; pi < 3; ++pi) {
            const int p = (pi == 2) ? 1 : 0;
            const int q = (pi == 0) ? 1 : 2;
            float apq = S[p][q];
            if (fabsf(apq) > 1e-15f) {
                float theta = (S[q][q] - S[p][p]) / (2.0f * apq);
                float t = ((theta >= 0.0f) ? 1.0f : -1.0f) /
                          (fabsf(theta) + sqrtf(theta * theta + 1.0f));
                float c0 = 1.0f / sqrtf(t * t + 1.0f);
                float s0 = t * c0;
                #pragma unroll
                for (int k = 0; k < 3; ++k) {
                    float skp = S[k][p], skq = S[k][q];
                    S[k][p] = c0 * skp - s0 * skq;
                    S[k][q] = s0 * skp + c0 * skq;
                }
                #pragma unroll
                for (int k = 0; k < 3; ++k) {
                    float spk = S[p][k], sqk = S[q][k];
                    S[p][k] = c0 * spk - s0 * sqk;
                    S[q][k] = s0 * spk + c0 * sqk;
                }
                #pragma unroll
                for (int k = 0; k < 3; ++k) {
                    float vkp = V[k][p], vkq = V[k][q];
                    V[k][p] = c0 * vkp - s0 * vkq;
                    V[k][q] = s0 * vkp + c0 * vkq;
                }
            }
        }
    }
    // sort eigenvalues descending (with V columns)
    float e[3] = { S[0][0], S[1][1], S[2][2] };
    #pragma unroll
    for (int a0 = 0; a0 < 2; ++a0)
        #pragma unroll
        for (int b0 = 0; b0 < 2; ++b0)
            if (e[b0] < e[b0 + 1]) {
                float te = e[b0]; e[b0] = e[b0 + 1]; e[b0 + 1] = te;
                #pragma unroll
                for (int k = 0; k < 3; ++k) {
                    float tv = V[k][b0]; V[k][b0] = V[k][b0 + 1]; V[k][b0 + 1] = tv;
                }
            }
    // U columns = C v_k / sigma_k ; reflection sign from det(C)
    float U[3][3];
    #pragma unroll
    for (int k = 0; k < 3; ++k) {
        float sig = sqrtf(fmaxf(e[k], 0.0f));
        float is = 1.0f / fmaxf(sig, 1e-12f);
        #pragma unroll
        for (int i = 0; i < 3; ++i)
            U[i][k] = (C_[i][0] * V[0][k] + C_[i][1] * V[1][k] + C_[i][2] * V[2][k]) * is;
    }
    float detC = C_[0][0] * (C_[1][1] * C_[2][2] - C_[1][2] * C_[2][1])
               - C_[0][1] * (C_[1][0] * C_[2][2] - C_[1][2] * C_[2][0])
               + C_[0][2] * (C_[1][0] * C_[2][1] - C_[1][1] * C_[2][0]);
    float d = (detC >= 0.0f) ? 1.0f : -1.0f;

    float Ra[3][3];
    #pragma unroll
    for (int i = 0; i < 3; ++i)
        #pragma unroll
        for (int j = 0; j < 3; ++j)
            Ra[i][j] = U[i][0] * V[j][0] + U[i][1] * V[j][1] + d * U[i][2] * V[j][2];

    // fused: M = Rh @ Ra, T = muY - muX @ Ra
    if (lane == 0) {
        float* mt = MT + ((size_t)hand * Bn + b) * 12;
        #pragma unroll
        for (int k = 0; k < 3; ++k)
            #pragma unroll
            for (int j = 0; j < 3; ++j)
                mt[k * 3 + j] = Rh[k][0] * Ra[0][j] + Rh[k][1] * Ra[1][j] + Rh[k][2] * Ra[2][j];
        float muX[3] = { muX0, muX1, muX2 };
        float muY[3] = { muY0, muY1, muY2 };
        #pragma unroll
        for (int j = 0; j < 3; ++j)
            mt[9 + j] = muY[j] - (muX[0] * Ra[0][j] + muX[1] * Ra[1][j] + muX[2] * Ra[2][j]);
    }
}

// ---------------------------------------------------------------------------
// Kernel 2: bulk copy of smplx verts into output (float4 vectorized)
__global__ void copy4_kernel(const float4* __restrict__ in, float4* __restrict__ out, int n) {
    int i = blockIdx.x * blockDim.x + threadIdx.x;
    if (i < n) out[i] = in[i];
}

// ---------------------------------------------------------------------------
// Kernel 3: WMMA affine transform of 16 hand verts per wave + scatter.
//   A (16x4 f32) = [x,y,z,1] per vertex row; B (4x16) = [M rows 0..2; T] in cols 0..2.
//   D = A*B : column j (<3) of row m is transformed coordinate j of vertex m.
__global__ void wmma_scatter_kernel(const float* __restrict__ rvert,
                                    const float* __restrict__ lvert,
                                    const int* __restrict__ ridx,
                                    const int* __restrict__ lidx,
                                    const float* __restrict__ MT,
                                    float* __restrict__ out) {
    const int tile = blockIdx.x;
    const int b = blockIdx.y;
    const int hand = blockIdx.z;
    const int lane = threadIdx.x;

    const float* vin = hand ? lvert : rvert;
    const int* idx = hand ? lidx : ridx;
    const float* mt = MT + ((size_t)hand * Bn + b) * 12;

    // A matrix: lanes 0-15 hold row M=lane, K=0,1 ; lanes 16-31 hold row M=lane-16, K=2,3
    int row = tile * 16 + (lane & 15);
    int rc = row < MVV ? row : (MVV - 1);
    const float* vp = vin + ((size_t)b * MVV + rc) * 3;
    v2f a;
    a.x = (lane < 16) ? vp[0] : vp[2];
    a.y = (lane < 16) ? vp[1] : 1.0f;

    // B matrix: VGPR0 rows K=0 (lanes 0-15) / K=2 (lanes 16-31); VGPR1 rows K=1 / K=3(=T)
    int n = lane & 15;
    v2f bm;
    float b0 = 0.0f, b1 = 0.0f;
    if (n < 3) {
        b0 = (lane < 16) ? mt[0 * 3 + n] : mt[2 * 3 + n];
        b1 = (lane < 16) ? mt[1 * 3 + n] : mt[9 + n];
    }
    bm.x = b0; bm.y = b1;

    v8f cz = {};
    // D = A x B + 0 : one v_wmma_f32_16x16x4_f32, full EXEC, wave32
    v8f dacc = __builtin_amdgcn_wmma_f32_16x16x4_f32(
        false, a, false, bm, (short)0, cz, false, false);

    // D layout: VGPR r -> row M = r (lanes 0-15) or r+8 (lanes 16-31), col N = lane&15
    if (n < 3) {
        int mbase = tile * 16 + ((lane < 16) ? 0 : 8);
        #pragma unroll
        for (int r = 0; r < 8; ++r) {
            int mrow = mbase + r;
            if (mrow < MVV) {
                int dst = idx[mrow];
                out[((size_t)b * NVV + dst) * 3 + n] = dacc[r];
            }
        }
    }
}

// ---------------------------------------------------------------------------
// Kernel 4a: Laplacian smoothing update into scratch (Jacobi semantics)
__global__ void smooth_gather_kernel(const float* __restrict__ v,
                                     const int* __restrict__ bidx,
                                     const int* __restrict__ nbr,
                                     float* __restrict__ upd) {
    int t = blockIdx.x * blockDim.x + threadIdx.x;
    if (t >= Bn * NBB) return;
    int b = t / NBB;
    int i = t - b * NBB;
    int vi = bidx[i];
    const int* row = nbr + (size_t)vi * KNN;
    const float* vb = v + (size_t)b * NVV * 3;
    float sx = 0.f, sy = 0.f, sz = 0.f;
    int cnt = 0;
    #pragma unroll
    for (int k = 0; k < KNN; ++k) {
        int nb_ = row[k];
        int valid = (nb_ != -1);
        int safe = valid ? nb_ : 0;
        const float* p = vb + (size_t)safe * 3;
        float w = valid ? 1.0f : 0.0f;
        sx += w * p[0]; sy += w * p[1]; sz += w * p[2];
        cnt += valid;
    }
    float invn = 1.0f / (float)(cnt > 0 ? cnt : 1);
    const float* cp = vb + (size_t)vi * 3;
    float* o = upd + (size_t)t * 3;
    o[0] = 0.5f * cp[0] + 0.5f * sx * invn;
    o[1] = 0.5f * cp[1] + 0.5f * sy * invn;
    o[2] = 0.5f * cp[2] + 0.5f * sz * invn;
}

// Kernel 4b: scatter updates back into v
__global__ void smooth_scatter_kernel(const float* __restrict__ upd,
                                      const int* __restrict__ bidx,
                                      float* __restrict__ v) {
    int t = blockIdx.x * blockDim.x + threadIdx.x;
    if (t >= Bn * NBB) return;
    int b = t / NBB;
    int i = t - b * NBB;
    int vi = bidx[i];
    float* o = v + ((size_t)b * NVV + vi) * 3;
    const float* u = upd + (size_t)t * 3;
    o[0] = u[0]; o[1] = u[1]; o[2] = u[2];
}

// ---------------------------------------------------------------------------
extern "C" void kernel_launch(void* const* d_in, const int* in_sizes, int n_in,
                              void* d_out, int out_size, void* d_ws, size_t ws_size,
                              hipStream_t stream) {
    const float* smplx_vert = (const float*)d_in[0];
    const float* smplx_kpt  = (const float*)d_in[1];
    const float* rvert = (const float*)d_in[2];
    const float* rkpt  = (const float*)d_in[3];
    const float* rpose = (const float*)d_in[4];
    const float* lvert = (const float*)d_in[5];
    const float* lkpt  = (const float*)d_in[6];
    const float* lpose = (const float*)d_in[7];
    const int* rigid_align_idx = (const int*)d_in[8];
    const int* r_align_smplx   = (const int*)d_in[9];
    const int* l_align_smplx   = (const int*)d_in[10];
    const int* ridx = (const int*)d_in[11];
    const int* lidx = (const int*)d_in[12];
    // d_in[13], d_in[14] (kpt scatter idx) unused: reference returns verts only
    const int* bidx = (const int*)d_in[15];
    const int* nbr  = (const int*)d_in[16];

    float* out = (float*)d_out;
    float* MT  = (float*)d_ws;                           // 2*256*12 floats = 24 KB
    float* upd = (float*)((char*)d_ws + 32768);          // 256*8192*3 floats ~ 24 MB

    // 1) per-(batch,hand) fused affine transform
    xform_kernel<<<dim3(Bn, 2), 32, 0, stream>>>(
        smplx_kpt, rkpt, rpose, lkpt, lpose,
        rigid_align_idx, r_align_smplx, l_align_smplx, MT);

    // 2) bulk copy body verts
    int n4 = (Bn * NVV * 3) / 4;   // 8,044,800 floats -> 2,011,200 float4
    copy4_kernel<<<(n4 + 255) / 256, 256, 0, stream>>>(
        (const float4*)smplx_vert, (float4*)out, n4);

    // 3) WMMA transform + scatter of hand verts (49 tiles of 16 verts)
    wmma_scatter_kernel<<<dim3((MVV + 15) / 16, Bn, 2), 32, 0, stream>>>(
        rvert, lvert, ridx, lidx, MT, out);

    // 4) 5 smoothing iterations, Jacobi two-phase
    int nt = Bn * NBB;
    int blocks = (nt + 255) / 256;
    for (int it = 0; it < 5; ++it) {
        smooth_gather_kernel<<<blocks, 256, 0, stream>>>(out, bidx, nbr, upd);
        smooth_scatter_kernel<<<blocks, 256, 0, stream>>>(upd, bidx, out);
    }
}